// RNN_4861902979716
// MI455X (gfx1250) — compile-verified
//
#include <hip/hip_runtime.h>

// Fused 3-layer tanh RNN for MI455X (gfx1250).
// T=2048, B=4096, H=4, L=3. Single pass: read x once, write out once.
//
// WMMA mapping: D(16x16) = A(W, 16x4) x B(batch columns, 4x16) + C(bias).
//   A (32-bit 16x4): lane m holds row m; VGPR0 = K0(lanes0-15)/K2(lanes16-31),
//     VGPR1 = K1/K3. Weight rows are stored TWICE: rows 0-3 and rows 8-11.
//     Because D places row M in VGPR M%8 (lanes 16-31 carry M>=8), the
//     duplication makes D VGPR r hold row r on ALL 32 lanes -> the B-layout
//     hidden state is rebuilt with two v_cndmask ops, no ds_swizzle, and only
//     2 tanh evaluations per layer.
//   B (32-bit 4x16): VGPR0 = K0(lo)/K2(hi), VGPR1 = K1/K3. A per-lane 8B load
//     of x[t][b0+lane%16][k0:k0+2] (k0 = lane<16?0:2) IS this layout, and the
//     output store is its mirror.

typedef float v2f __attribute__((ext_vector_type(2)));
typedef float v8f __attribute__((ext_vector_type(8)));

#define T_STEPS 2048
#define BATCH   4096
#define HID     4
#define LAYERS  3

__device__ __forceinline__ float fast_tanh(float x) {
  // tanh(x) = 1 - 2/(1 + exp(2x)) ; exp(2x) = exp2(x * 2*log2(e))
  float e = __builtin_amdgcn_exp2f(x * 2.885390081777927f);
  float r = __builtin_amdgcn_rcpf(e + 1.0f);
  return __builtin_fmaf(-2.0f, r, 1.0f);
}

__global__ __launch_bounds__(256) void rnn3_fused_wmma(
    const float* __restrict__ x, const float* __restrict__ h0,
    const float* __restrict__ Wih, const float* __restrict__ Whh,
    const float* __restrict__ bih, const float* __restrict__ bhh,
    float* __restrict__ out)
{
  const int lane   = threadIdx.x & 31;
  const int waveId = (blockIdx.x * blockDim.x + threadIdx.x) >> 5;
  const int b0     = waveId * 16;          // 16 batch elements per wave
  const int m      = lane & 15;            // A row / B-D batch column
  const int k0     = (lane >> 4) * 2;      // K pair handled by this half-wave
  const bool lo    = (lane < 16);

  // ---- Setup: weights into A layout (rows duplicated at +8), bias into C,
  //      h0 into B layout ----
  v2f a_ih[LAYERS], a_hh[LAYERS], bh[LAYERS];
  v8f cbias[LAYERS];
#pragma unroll
  for (int l = 0; l < LAYERS; ++l) {
    const int r = m & 7;                    // rows 0-3 and 8-11 carry W[r]
    float w0 = 0.f, w1 = 0.f, u0 = 0.f, u1 = 0.f;
    if (r < HID) {
      const int base = (l * HID + r) * HID + k0;
      w0 = Wih[base];  w1 = Wih[base + 1];
      u0 = Whh[base];  u1 = Whh[base + 1];
    }
    a_ih[l].x = w0; a_ih[l].y = w1;
    a_hh[l].x = u0; a_hh[l].y = u1;

    v8f c = {};                             // broadcast bias: also correct for
    c[0] = bih[l * HID + 0] + bhh[l * HID + 0];   // the duplicated rows 8-11
    c[1] = bih[l * HID + 1] + bhh[l * HID + 1];
    c[2] = bih[l * HID + 2] + bhh[l * HID + 2];
    c[3] = bih[l * HID + 3] + bhh[l * HID + 3];
    cbias[l] = c;

    const float* hp = h0 + ((size_t)l * BATCH + (b0 + m)) * HID + k0;
    bh[l].x = hp[0]; bh[l].y = hp[1];       // initial hidden state, B layout
  }

  // ---- Streaming pointers (per-lane 8B slices, already in B layout) ----
  const size_t strideV2 = (size_t)BATCH * HID / 2;   // v2f elems per timestep
  const v2f* xp = (const v2f*)(x   + (size_t)(b0 + m) * HID + k0);
  v2f*       op = (v2f*)      (out + (size_t)(b0 + m) * HID + k0);

  v2f xnext = xp[0];                        // software-pipelined x stream
#pragma unroll 2
  for (int t = 0; t < T_STEPS; ++t) {
    v2f bin = xnext;
    const int tn = (t + 1 < T_STEPS) ? t + 1 : t;
    xnext = xp[(size_t)tn * strideV2];
    const int tp = (t + 8 < T_STEPS) ? t + 8 : t;
    __builtin_prefetch((const void*)(xp + (size_t)tp * strideV2), 0, 3);

#pragma unroll
    for (int l = 0; l < LAYERS; ++l) {
      // D = Whh*h + bias (old operand first), then D += Wih*in (fresh operand)
      v8f acc = __builtin_amdgcn_wmma_f32_16x16x4_f32(
          false, a_hh[l], false, bh[l], (short)0, cbias[l], false, false);
      acc = __builtin_amdgcn_wmma_f32_16x16x4_f32(
          false, a_ih[l], false, bin, (short)0, acc, false, false);

      // rows r and r+8 are identical -> select straight into B layout,
      // then only 2 activations per layer
      float p0 = lo ? acc[0] : acc[2];
      float p1 = lo ? acc[1] : acc[3];
      v2f nh;
      nh.x = fast_tanh(p0);
      nh.y = fast_tanh(p1);
      bh[l] = nh;
      bin   = nh;                           // feeds next layer this timestep
    }
    op[(size_t)t * strideV2] = bin;         // last layer's h_t, coalesced b64
  }
}

extern "C" void kernel_launch(void* const* d_in, const int* in_sizes, int n_in,
                              void* d_out, int out_size, void* d_ws, size_t ws_size,
                              hipStream_t stream) {
  const float* x   = (const float*)d_in[0];
  const float* h   = (const float*)d_in[1];
  const float* Wih = (const float*)d_in[2];
  const float* Whh = (const float*)d_in[3];
  const float* bih = (const float*)d_in[4];
  const float* bhh = (const float*)d_in[5];
  float* out = (float*)d_out;

  // 16 batches per wave, 8 waves per block -> 128 batches/block, 32 blocks.
  dim3 block(256);
  dim3 grid(BATCH / 128);
  rnn3_fused_wmma<<<grid, block, 0, stream>>>(x, h, Wih, Whh, bih, bhh, out);
}